// PraxisMemory_42073499631739
// MI455X (gfx1250) — compile-verified
//
#include <hip/hip_runtime.h>
#include <hip/hip_bf16.h>

// ---------------- problem constants (from reference) ----------------
#define Bdim     2
#define Sdim     1024
#define Hdim     1024
#define Vdim     32000
#define NUM_MEM  256
#define MAX_LEN  16
#define K_SIM    8
#define K_CONT   4
#define WIN      20
#define GAMMA_C  2.0f
#define MEMLEN   ((K_SIM + K_CONT) * MAX_LEN)   // 192
#define CTXLEN   (MEMLEN + Sdim)                // 1216
#define NSEG     (Bdim * Sdim)                  // 2048
#define NROWS    (Bdim * K_SIM * MAX_LEN + K_CONT * MAX_LEN) // 320 storage rows

// ---- logits GEMM tiling: 64M x 128N block tiles, K staged in 32-panels ----
#define BM   64
#define BN   128
#define KC   32
#define NBM  (NSEG / BM)      // 32
#define NBN  (Vdim / BN)      // 250
#define NPART (NBM * NBN * BM) // 512000 partial rows

typedef float v2f __attribute__((ext_vector_type(2)));
typedef float v8f __attribute__((ext_vector_type(8)));

// ================================================================
// Kernel 1: blocked fp32 WMMA GEMM (query @ W_brain) with online
// logsumexp + target-logit capture. 256 threads = 8 waves,
// each wave owns a 16x64 strip (4 WMMA accumulators).
// ================================================================
__global__ __launch_bounds__(256)
void k_surprise_partial(const float* __restrict__ q, const float* __restrict__ Wb,
                        const int* __restrict__ tgt,
                        float* __restrict__ pm, float* __restrict__ ps,
                        float* __restrict__ pt)
{
    __shared__ float As[BM][KC + 4];          // 64 x 36  (pad kills bank conflicts)
    __shared__ float Bs[KC][BN + 4];          // 32 x 132
    __shared__ float rm[2][BM], rs[2][BM], rt[2][BM];

    const int mb = blockIdx.x, nb = blockIdx.y;
    const int m0 = mb * BM,    n0 = nb * BN;
    const int tid  = threadIdx.x;
    const int w    = tid >> 5, lane = tid & 31;
    const int wm   = w >> 1,   wn   = w & 1;   // 4 x 2 wave grid
    const int half = lane >> 4, l16 = lane & 15;

    v8f c[4];
    for (int s = 0; s < 4; ++s) c[s] = (v8f){0.f,0.f,0.f,0.f,0.f,0.f,0.f,0.f};

    int tg[8];
    const int rowbase = m0 + wm * 16 + 8 * half;
    for (int i = 0; i < 8; ++i) tg[i] = tgt[rowbase + i];

    for (int kc = 0; kc < Hdim; kc += KC) {
        __syncthreads();
        // stage A panel: 64x32 floats = 512 float4, 2 per thread (coalesced b128)
        for (int it = 0; it < 2; ++it) {
            const int i  = tid + it * 256;     // 0..511
            const int r  = i >> 3, c4 = i & 7;
            float4 v = *(const float4*)(q + (size_t)(m0 + r) * Hdim + kc + c4 * 4);
            *(float4*)(&As[r][c4 * 4]) = v;
        }
        // stage B panel: 32x128 floats = 1024 float4, 4 per thread (coalesced b128)
        for (int it = 0; it < 4; ++it) {
            const int i  = tid + it * 256;     // 0..1023
            const int r  = i >> 5, c4 = i & 31;
            float4 v = *(const float4*)(Wb + (size_t)(kc + r) * Vdim + n0 + c4 * 4);
            *(float4*)(&Bs[r][c4 * 4]) = v;
        }
        __syncthreads();
        // 8 k-steps x 4 sub-tiles = 32 WMMAs per staged panel
        for (int ks = 0; ks < KC; ks += 4) {
            v2f a;
            a.x = As[wm * 16 + l16][ks + 2 * half];
            a.y = As[wm * 16 + l16][ks + 2 * half + 1];
            for (int s = 0; s < 4; ++s) {
                v2f b;
                const int col = wn * 64 + s * 16 + l16;
                b.x = Bs[ks + 2 * half][col];
                b.y = Bs[ks + 2 * half + 1][col];
                c[s] = __builtin_amdgcn_wmma_f32_16x16x4_f32(false, a, false, b,
                                                             (short)0, c[s], false, false);
            }
        }
    }

    // per-lane stats over the 4 columns this lane owns for each of 8 rows
    float mx[8], sm[8], tp[8];
    for (int i = 0; i < 8; ++i) {
        float m = c[0][i];
        for (int s = 1; s < 4; ++s) m = fmaxf(m, c[s][i]);
        float ssum = 0.f, t = 0.f;
        for (int s = 0; s < 4; ++s) {
            const float v = c[s][i];
            ssum += __expf(v - m);
            if (n0 + wn * 64 + s * 16 + l16 == tg[i]) t += v;
        }
        mx[i] = m; sm[i] = ssum; tp[i] = t;
    }
    // combine across the 16 lanes of each half-wave
    for (int off = 8; off >= 1; off >>= 1) {
        for (int i = 0; i < 8; ++i) {
            const float om = __shfl_xor(mx[i], off, 16);
            const float os = __shfl_xor(sm[i], off, 16);
            const float ot = __shfl_xor(tp[i], off, 16);
            if (om > mx[i]) { sm[i] = sm[i] * __expf(mx[i] - om) + os; mx[i] = om; }
            else            { sm[i] += os * __expf(om - mx[i]); }
            tp[i] += ot;
        }
    }
    __syncthreads();
    if (l16 == 0) {
        for (int i = 0; i < 8; ++i) {
            const int row = wm * 16 + 8 * half + i;
            rm[wn][row] = mx[i]; rs[wn][row] = sm[i]; rt[wn][row] = tp[i];
        }
    }
    __syncthreads();
    // combine the two N-halves, write per-(block,row) partials
    if (tid < BM) {
        const float m0v = rm[0][tid], s0v = rs[0][tid];
        const float m1v = rm[1][tid], s1v = rs[1][tid];
        float m, ssum;
        if (m1v > m0v) { m = m1v; ssum = s0v * __expf(m0v - m1v) + s1v; }
        else           { m = m0v; ssum = s0v + s1v * __expf(m1v - m0v); }
        const size_t idx = ((size_t)mb * NBN + nb) * BM + tid;
        pm[idx] = m; ps[idx] = ssum; pt[idx] = rt[0][tid] + rt[1][tid];
    }
}

// ================================================================
// Kernel 2: combine per-N-block partials -> surprise[b,s]
// ================================================================
__global__ void k_surprise_combine(const float* __restrict__ pm,
                                   const float* __restrict__ ps,
                                   const float* __restrict__ pt,
                                   float* __restrict__ sur)
{
    const int r = blockIdx.x * blockDim.x + threadIdx.x;
    if (r >= NSEG) return;
    const int mb = r / BM, rowin = r % BM;
    float m = -3.0e38f, s = 0.f, t = 0.f;
    for (int c = 0; c < NBN; ++c) {
        const size_t idx = ((size_t)mb * NBN + c) * BM + rowin;
        const float om = pm[idx], os = ps[idx];
        if (om > m) { s = s * __expf(m - om) + os; m = om; }
        else        { s += os * __expf(om - m); }
        t += pt[idx];
    }
    sur[r] = __logf(s) + m - t;                  // logsumexp - logit[target]
}

// ================================================================
// Kernel 3: windowed threshold -> boundaries -> event ids (cumsum)
// ================================================================
__global__ void k_boundaries(const float* __restrict__ sur, int* __restrict__ ev)
{
    __shared__ unsigned char bf[Sdim];
    const int b = blockIdx.x;
    const float* sb = sur + (size_t)b * Sdim;
    for (int s = threadIdx.x; s < Sdim; s += blockDim.x) {
        const int i0 = (s >= WIN - 1) ? (s - (WIN - 1)) : 0;
        float mean = 0.f;
        for (int j = 0; j < WIN; ++j) mean += sb[i0 + j];
        mean *= (1.f / WIN);
        float var = 0.f;
        for (int j = 0; j < WIN; ++j) { float d = sb[i0 + j] - mean; var += d * d; }
        var *= (1.f / (WIN - 1));
        const float thr = mean + GAMMA_C * sqrtf(var);
        bf[s] = (sb[s] > thr) ? 1 : 0;
    }
    __syncthreads();
    if (threadIdx.x == 0) {
        int run = 0;
        for (int s = 0; s < Sdim; ++s) {
            if (s > 0) run += bf[s - 1];
            ev[b * Sdim + s] = run + b * Sdim;
        }
    }
}

// ================================================================
// Kernels 4/5: per-event stats (count, start) and event mean key
// ================================================================
__global__ void k_init(int* __restrict__ cnt, int* __restrict__ segstart)
{
    const int i = blockIdx.x * blockDim.x + threadIdx.x;
    if (i < NSEG) { cnt[i] = 0; segstart[i] = 0x7fffffff; }
}

__global__ void k_token_stats(const int* __restrict__ ev, int* __restrict__ cnt,
                              int* __restrict__ segstart)
{
    const int fi = blockIdx.x * blockDim.x + threadIdx.x;
    if (fi >= NSEG) return;
    const int e = ev[fi];
    atomicAdd(&cnt[e], 1);
    atomicMin(&segstart[e], fi % Sdim);
}

__global__ void k_ev_keys(const float* __restrict__ q, const int* __restrict__ cnt,
                          const int* __restrict__ segstart, float* __restrict__ evk)
{
    const int e = blockIdx.x;
    const int c = cnt[e];
    if (c == 0) return;
    const int b = e / Sdim;
    const int s0 = segstart[e];
    const float* base = q + ((size_t)b * Sdim + s0) * Hdim;
    const float inv = 1.f / (float)c;
    for (int d = threadIdx.x; d < Hdim; d += blockDim.x) {
        float acc = 0.f;
        for (int j = 0; j < c; ++j) acc += base[(size_t)j * Hdim + d];
        evk[(size_t)e * Hdim + d] = acc * inv;
    }
}

// ================================================================
// Kernel 6: serial rank / slot / chosen / cont_idx bookkeeping
// ================================================================
__global__ void k_rank(const int* __restrict__ cnt, int* __restrict__ rank_of,
                       int* __restrict__ chosen, int* __restrict__ slot_rank,
                       int* __restrict__ cont_idx)
{
    if (threadIdx.x != 0 || blockIdx.x != 0) return;
    for (int s = 0; s < NUM_MEM; ++s) { chosen[s] = -1; slot_rank[s] = -1; }
    int R = 0;
    for (int e = 0; e < NSEG; ++e) {
        if (cnt[e] > 0) {
            rank_of[e] = R;
            const int sl = R % NUM_MEM;
            chosen[sl] = e;               // latest (max rank) event wins
            slot_rank[sl] = R;
            ++R;
        } else rank_of[e] = -1;
    }
    // top-4 of slot_rank, stable (lowest index first among ties)
    int picked[K_CONT];
    for (int j = 0; j < K_CONT; ++j) {
        int best = -2147483647, bi = 0;
        for (int s = 0; s < NUM_MEM; ++s) {
            bool skip = false;
            for (int p = 0; p < j; ++p) if (picked[p] == s) skip = true;
            if (!skip && slot_rank[s] > best) { best = slot_rank[s]; bi = s; }
        }
        picked[j] = bi;
        cont_idx[j] = bi;
    }
}

__global__ void k_memory_keys(const int* __restrict__ chosen,
                              const int* __restrict__ slot_rank,
                              const float* __restrict__ evk, float* __restrict__ mk)
{
    const int sl = blockIdx.x;
    const bool valid = slot_rank[sl] >= 0;
    const int e = valid ? chosen[sl] : 0;
    for (int d = threadIdx.x; d < Hdim; d += blockDim.x)
        mk[(size_t)sl * Hdim + d] = valid ? evk[(size_t)e * Hdim + d] : 0.f;
}

__global__ void k_memvals_scatter(const float* __restrict__ q, const int* __restrict__ ev,
                                  const int* __restrict__ rank_of,
                                  const int* __restrict__ chosen,
                                  const int* __restrict__ segstart,
                                  float* __restrict__ mv)
{
    const int fi = blockIdx.x;
    const int e = ev[fi];
    const int r = rank_of[e];
    if (r < 0) return;
    const int sl = r % NUM_MEM;
    if (chosen[sl] != e) return;
    const int off = (fi % Sdim) - segstart[e];
    if (off < 0 || off >= MAX_LEN) return;
    float* dst = mv + ((size_t)sl * MAX_LEN + off) * Hdim;
    const float* src = q + (size_t)fi * Hdim;
    for (int d = threadIdx.x; d < Hdim; d += blockDim.x) dst[d] = src[d];
}

// ================================================================
// q_mean path (linearity: mean over S first, then one matvec)
// ================================================================
__global__ void k_qbar(const float* __restrict__ q, float* __restrict__ qbar)
{
    const int b = blockIdx.x;
    const int h = blockIdx.y * blockDim.x + threadIdx.x;
    float acc = 0.f;
    const float* base = q + (size_t)b * Sdim * Hdim + h;
    for (int s = 0; s < Sdim; ++s) acc += base[(size_t)s * Hdim];
    qbar[b * Hdim + h] = acc * (1.f / Sdim);
}

__global__ void k_qmean(const float* __restrict__ qbar, const float* __restrict__ Wsim,
                        const float* __restrict__ bsim, float* __restrict__ qm)
{
    const int idx = blockIdx.x * blockDim.x + threadIdx.x;
    if (idx >= Bdim * Hdim) return;
    const int b = idx / Hdim, n = idx % Hdim;
    float acc = bsim[n];
    for (int h = 0; h < Hdim; ++h) acc += qbar[b * Hdim + h] * Wsim[(size_t)h * Hdim + n];
    qm[idx] = acc;
}

__global__ void k_sims(const float* __restrict__ qm, const float* __restrict__ mk,
                       const int* __restrict__ slot_rank, float* __restrict__ sims)
{
    __shared__ float red[256];
    const int b = blockIdx.x / NUM_MEM, sl = blockIdx.x % NUM_MEM;
    float acc = 0.f;
    const bool valid = slot_rank[sl] >= 0;
    if (valid)
        for (int d = threadIdx.x; d < Hdim; d += blockDim.x)
            acc += qm[b * Hdim + d] * mk[(size_t)sl * Hdim + d];
    red[threadIdx.x] = acc;
    __syncthreads();
    for (int off = 128; off > 0; off >>= 1) {
        if (threadIdx.x < off) red[threadIdx.x] += red[threadIdx.x + off];
        __syncthreads();
    }
    if (threadIdx.x == 0) sims[b * NUM_MEM + sl] = valid ? red[0] : -1e30f;
}

// serial top-K_SIM per batch + build 320-row gather table
__global__ void k_topk_rows(const float* __restrict__ sims, const int* __restrict__ cont_idx,
                            int* __restrict__ rowsrc)
{
    if (threadIdx.x != 0 || blockIdx.x != 0) return;
    for (int b = 0; b < Bdim; ++b) {
        int pk[K_SIM];
        for (int j = 0; j < K_SIM; ++j) {
            float best = -3.4e38f; int bi = 0;
            for (int s = 0; s < NUM_MEM; ++s) {
                bool skip = false;
                for (int p = 0; p < j; ++p) if (pk[p] == s) skip = true;
                if (!skip && sims[b * NUM_MEM + s] > best) { best = sims[b * NUM_MEM + s]; bi = s; }
            }
            pk[j] = bi;
            for (int l = 0; l < MAX_LEN; ++l)
                rowsrc[b * K_SIM * MAX_LEN + j * MAX_LEN + l] = bi * MAX_LEN + l;
        }
    }
    for (int j = 0; j < K_CONT; ++j)
        for (int l = 0; l < MAX_LEN; ++l)
            rowsrc[Bdim * K_SIM * MAX_LEN + j * MAX_LEN + l] = cont_idx[j] * MAX_LEN + l;
}

__global__ void k_gather(const float* __restrict__ mv, const int* __restrict__ rowsrc,
                         float* __restrict__ X)
{
    const int r = blockIdx.x;
    const float* src = mv + (size_t)rowsrc[r] * Hdim;
    float* dst = X + (size_t)r * Hdim;
    for (int d = threadIdx.x; d < Hdim; d += blockDim.x) dst[d] = src[d];
}

// ================================================================
// Generic fp32 WMMA GEMM: C = A(MxK) @ B(KxN) + bias, optional ReLU
// one wave per 16x16 tile (problem here is tiny: 320x1024x1024)
// ================================================================
__global__ __launch_bounds__(32)
void k_gemm_wmma(const float* __restrict__ A, const float* __restrict__ Bm,
                 const float* __restrict__ bias, float* __restrict__ C,
                 int K, int N, int relu)
{
    const int m0 = blockIdx.x * 16, n0 = blockIdx.y * 16;
    const int lane = threadIdx.x, half = lane >> 4, l16 = lane & 15;
    v8f c = {0.f, 0.f, 0.f, 0.f, 0.f, 0.f, 0.f, 0.f};
    const float* Ar = A + (size_t)(m0 + l16) * K;
    const float* Bp = Bm + n0 + l16;
    for (int k0 = 0; k0 < K; k0 += 4) {
        v2f a, b;
        a.x = Ar[k0 + 2 * half];
        a.y = Ar[k0 + 2 * half + 1];
        b.x = Bp[(size_t)(k0 + 2 * half) * N];
        b.y = Bp[(size_t)(k0 + 2 * half + 1) * N];
        c = __builtin_amdgcn_wmma_f32_16x16x4_f32(false, a, false, b,
                                                  (short)0, c, false, false);
    }
    const float bv = bias[n0 + l16];
    for (int i = 0; i < 8; ++i) {
        float v = c[i] + bv;
        if (relu) v = fmaxf(v, 0.f);
        C[(size_t)(m0 + i + 8 * half) * N + n0 + l16] = v;
    }
}

// ================================================================
// Output assembly
// ================================================================
__global__ void k_context(const float* __restrict__ Y, const float* __restrict__ key,
                          float* __restrict__ o1, float* __restrict__ o2)
{
    const int r = blockIdx.x;          // 0..CTXLEN-1
    const int b = blockIdx.y;
    const float* src;
    if (r < K_SIM * MAX_LEN)                     // 0..127: storage(sim_buf)
        src = Y + (size_t)(b * K_SIM * MAX_LEN + r) * Hdim;
    else if (r < MEMLEN)                         // 128..191: storage(cont_buf)
        src = Y + (size_t)(Bdim * K_SIM * MAX_LEN + (r - K_SIM * MAX_LEN)) * Hdim;
    else                                         // 192..: key
        src = key + ((size_t)b * Sdim + (r - MEMLEN)) * Hdim;
    const size_t dofs = ((size_t)b * CTXLEN + r) * Hdim;
    for (int d = threadIdx.x; d < Hdim; d += blockDim.x) {
        const float v = src[d];
        o1[dofs + d] = v;
        o2[dofs + d] = v;
    }
}

__global__ void k_mask(const float* __restrict__ amask, float* __restrict__ om)
{
    const int idx = blockIdx.x * blockDim.x + threadIdx.x;
    if (idx >= Bdim * CTXLEN) return;
    const int b = idx / CTXLEN, p = idx % CTXLEN;
    om[idx] = (p < MEMLEN) ? 1.f : amask[b * Sdim + (p - MEMLEN)];
}

// ================================================================
// launcher
// ================================================================
extern "C" void kernel_launch(void* const* d_in, const int* in_sizes, int n_in,
                              void* d_out, int out_size, void* d_ws, size_t ws_size,
                              hipStream_t stream)
{
    (void)in_sizes; (void)n_in; (void)out_size; (void)ws_size;
    const float* q    = (const float*)d_in[0];
    const float* key  = (const float*)d_in[1];
    /* value d_in[2] unused by outputs */
    const float* am   = (const float*)d_in[3];
    const int*   tgt  = (const int*)  d_in[4];
    const float* Wb   = (const float*)d_in[5];
    const float* Ws1  = (const float*)d_in[6];
    const float* bs1  = (const float*)d_in[7];
    const float* Ws2  = (const float*)d_in[8];
    const float* bs2  = (const float*)d_in[9];
    const float* Wsim = (const float*)d_in[10];
    const float* bsim = (const float*)d_in[11];

    char* wp = (char*)d_ws;
    auto alloc = [&](size_t bytes) -> char* {
        char* p = wp;
        wp += (bytes + 255) & ~((size_t)255);
        return p;
    };
    float* pm       = (float*)alloc((size_t)NPART * 4);
    float* ps       = (float*)alloc((size_t)NPART * 4);
    float* pt       = (float*)alloc((size_t)NPART * 4);
    float* sur      = (float*)alloc((size_t)NSEG * 4);
    int*   ev       = (int*)  alloc((size_t)NSEG * 4);
    int*   cnt      = (int*)  alloc((size_t)NSEG * 4);
    int*   segstart = (int*)  alloc((size_t)NSEG * 4);
    int*   rank_of  = (int*)  alloc((size_t)NSEG * 4);
    int*   chosen   = (int*)  alloc((size_t)NUM_MEM * 4);
    int*   slotrank = (int*)  alloc((size_t)NUM_MEM * 4);
    int*   contidx  = (int*)  alloc((size_t)K_CONT * 4);
    int*   rowsrc   = (int*)  alloc((size_t)NROWS * 4);
    float* evk      = (float*)alloc((size_t)NSEG * Hdim * 4);
    float* mk       = (float*)alloc((size_t)NUM_MEM * Hdim * 4);
    float* mv       = (float*)alloc((size_t)NUM_MEM * MAX_LEN * Hdim * 4);
    float* qbar     = (float*)alloc((size_t)Bdim * Hdim * 4);
    float* qm       = (float*)alloc((size_t)Bdim * Hdim * 4);
    float* sims     = (float*)alloc((size_t)Bdim * NUM_MEM * 4);
    float* X        = (float*)alloc((size_t)NROWS * Hdim * 4);
    float* T        = (float*)alloc((size_t)NROWS * Hdim * 4);
    float* Y        = (float*)alloc((size_t)NROWS * Hdim * 4);

    float* out  = (float*)d_out;
    float* o_q  = out;
    float* o_c1 = o_q  + (size_t)Bdim * Sdim * Hdim;
    float* o_c2 = o_c1 + (size_t)Bdim * CTXLEN * Hdim;
    float* o_m  = o_c2 + (size_t)Bdim * CTXLEN * Hdim;

    // 1) surprise via blocked WMMA GEMM + online logsumexp
    k_surprise_partial<<<dim3(NBM, NBN), 256, 0, stream>>>(q, Wb, tgt, pm, ps, pt);
    k_surprise_combine<<<(NSEG + 255) / 256, 256, 0, stream>>>(pm, ps, pt, sur);

    // 2) boundaries -> event ids
    k_boundaries<<<Bdim, 256, 0, stream>>>(sur, ev);

    // 3) segment stats + event keys
    k_init<<<(NSEG + 255) / 256, 256, 0, stream>>>(cnt, segstart);
    k_token_stats<<<(NSEG + 255) / 256, 256, 0, stream>>>(ev, cnt, segstart);
    k_ev_keys<<<NSEG, 256, 0, stream>>>(q, cnt, segstart, evk);

    // 4) slot bookkeeping + memory keys/values
    k_rank<<<1, 32, 0, stream>>>(cnt, rank_of, chosen, slotrank, contidx);
    k_memory_keys<<<NUM_MEM, 256, 0, stream>>>(chosen, slotrank, evk, mk);
    hipMemsetAsync(mv, 0, (size_t)NUM_MEM * MAX_LEN * Hdim * 4, stream);
    k_memvals_scatter<<<NSEG, 256, 0, stream>>>(q, ev, rank_of, chosen, segstart, mv);

    // 5) similarity selection
    k_qbar<<<dim3(Bdim, Hdim / 256), 256, 0, stream>>>(q, qbar);
    k_qmean<<<(Bdim * Hdim + 255) / 256, 256, 0, stream>>>(qbar, Wsim, bsim, qm);
    k_sims<<<Bdim * NUM_MEM, 256, 0, stream>>>(qm, mk, slotrank, sims);
    k_topk_rows<<<1, 32, 0, stream>>>(sims, contidx, rowsrc);
    k_gather<<<NROWS, 256, 0, stream>>>(mv, rowsrc, X);

    // 6) storage MLP: two WMMA GEMMs (relu fused into first)
    k_gemm_wmma<<<dim3(NROWS / 16, Hdim / 16), 32, 0, stream>>>(X, Ws1, bs1, T, Hdim, Hdim, 1);
    k_gemm_wmma<<<dim3(NROWS / 16, Hdim / 16), 32, 0, stream>>>(T, Ws2, bs2, Y, Hdim, Hdim, 0);

    // 7) assemble outputs: (query, context, context, new_mask)
    hipMemcpyAsync(o_q, q, (size_t)Bdim * Sdim * Hdim * 4, hipMemcpyDeviceToDevice, stream);
    k_context<<<dim3(CTXLEN, Bdim), 256, 0, stream>>>(Y, key, o_c1, o_c2);
    k_mask<<<(Bdim * CTXLEN + 255) / 256, 256, 0, stream>>>(am, o_m);
}